// MeshTokenizer_81793357185181
// MI455X (gfx1250) — compile-verified
//
#include <hip/hip_runtime.h>
#include <hip/hip_bf16.h>
#include <stdint.h>

#ifndef __has_builtin
#define __has_builtin(x) 0
#endif

#define AS1 __attribute__((address_space(1)))
#define AS3 __attribute__((address_space(3)))

typedef int g4i __attribute__((vector_size(16)));   // matches builtin param: int __vector(4)

// ---------------- problem constants (match reference) ----------------
#define B_      64
#define NV_     8192
#define NF_     16384
#define ROW_    (NF_ * 10 + 1)               // 163841 tokens per batch row
#define L_IDS   ((size_t)B_ * ROW_)          // input_ids / attention_mask elems
#define C_CODES ((size_t)B_ * NF_ * 9)       // codes / discrete_face_coords elems

// ---------------- tiling ----------------
#define THREADS          256                 // 8 wave32 waves
#define FACES_PER_BLOCK  2048
#define TILES            (NF_ / FACES_PER_BLOCK)   // 8
#define LDS_VERT_BYTES   (NV_ * 3 * 4)             // 96 KB  (one batch's vertices)
#define LDS_CODE_BYTES   (FACES_PER_BLOCK * 9)     // 18 KB  (u8 codes tile)
#define SH_BYTES         (LDS_VERT_BYTES + LDS_CODE_BYTES)

static_assert(LDS_VERT_BYTES % (THREADS * 16) == 0, "b128 staging must tile evenly");
#define STAGE_ITERS (LDS_VERT_BYTES / (THREADS * 16))   // 24 async b128 per thread

// ---------------- CDNA5 async global->LDS helpers ----------------
static __device__ __forceinline__ void async_g2l_b128(const void* g, void* l) {
#if __has_builtin(__builtin_amdgcn_global_load_async_to_lds_b128)
    __builtin_amdgcn_global_load_async_to_lds_b128((AS1 g4i*)g, (AS3 g4i*)l, 0, 0);
#else
    unsigned loff = (unsigned)(unsigned long long)(uintptr_t)l;   // flat LDS addr low 32b
    unsigned long long ga = (unsigned long long)(uintptr_t)g;
    asm volatile("global_load_async_to_lds_b128 %0, %1, off"
                 :: "v"(loff), "v"(ga) : "memory");
#endif
}

static __device__ __forceinline__ void wait_async0() {
#if __has_builtin(__builtin_amdgcn_s_wait_asynccnt)
    __builtin_amdgcn_s_wait_asynccnt(0);
#else
    asm volatile("s_wait_asynccnt 0" ::: "memory");
#endif
}

// discretize: q = (t+1)/2*128 - 0.5 = t*64 + 63.5 ; round-half-even ; clamp [0,127]
static __device__ __forceinline__ float quantize(float t) {
    float q = rintf(t * 64.0f + 63.5f);           // v_rndne_f32 (RTE) == jnp.round
    return fminf(127.0f, fmaxf(0.0f, q));
}

// ---------------- kernel 1: attention_mask fill + input_ids pads ----------------
__global__ __launch_bounds__(THREADS) void mesh_fill_mask_pads(float* __restrict__ out) {
    size_t gid = (size_t)blockIdx.x * blockDim.x + threadIdx.x;
    const size_t n4 = L_IDS / 4;                  // L_IDS divisible by 4
    if (gid < n4) {
        float4 ones = make_float4(1.0f, 1.0f, 1.0f, 1.0f);
        reinterpret_cast<float4*>(out + L_IDS)[gid] = ones;   // attention_mask
    }
    if (gid < B_) {                               // input_ids pads (-1 at both ends)
        out[gid * (size_t)ROW_]            = -1.0f;
        out[gid * (size_t)ROW_ + ROW_ - 1] = -1.0f;
    }
}

// ---------------- kernel 2: gather + quantize + pack tokens ----------------
__global__ __launch_bounds__(THREADS) void mesh_tokenize_kernel(
        const float* __restrict__ vertices,       // (B, NV, 3)
        const int*   __restrict__ faces,          // (B, NF, 3)
        float*       __restrict__ out)
{
    extern __shared__ char smem[];
    float*         sv = reinterpret_cast<float*>(smem);                          // 96 KB verts
    unsigned char* sc = reinterpret_cast<unsigned char*>(smem + LDS_VERT_BYTES); // 18 KB codes

    const int tid   = threadIdx.x;
    const int b     = blockIdx.x / TILES;
    const int fbase = (blockIdx.x % TILES) * FACES_PER_BLOCK;   // face index within batch

    const int* fptr = faces + ((size_t)b * NF_ + fbase) * 3;

    // CDNA5 global_prefetch_b8: warm the face-index tile (2048*12B) while verts stage
    __builtin_prefetch(fptr + tid * 24, 0, 0);

    // Stage this batch's entire vertex array into LDS with async b128 copies.
    const char* gsrc = reinterpret_cast<const char*>(vertices + (size_t)b * NV_ * 3);
    char*       ldst = reinterpret_cast<char*>(sv);
    #pragma unroll
    for (int i = 0; i < STAGE_ITERS; ++i) {
        unsigned off = (unsigned)tid * 16u + (unsigned)i * (THREADS * 16u);
        async_g2l_b128(gsrc + off, ldst + off);
    }
    wait_async0();
    __syncthreads();

    // Phase A: quantize 8 faces per thread into the u8 LDS code tile.
    #pragma unroll
    for (int i = 0; i < FACES_PER_BLOCK / THREADS; ++i) {
        int f = tid + i * THREADS;
        const int* fp = fptr + f * 3;
        int idx0 = fp[0], idx1 = fp[1], idx2 = fp[2];
        int idx[3] = { idx0, idx1, idx2 };
        #pragma unroll
        for (int v = 0; v < 3; ++v) {
            int vb = idx[v] * 3;
            #pragma unroll
            for (int c = 0; c < 3; ++c) {
                sc[f * 9 + v * 3 + c] = (unsigned char)quantize(sv[vb + c]);
            }
        }
    }
    __syncthreads();

    // Phase B1: codes + discrete_face_coords, coalesced float4 stores.
    {
        float* out_codes = out + 2 * L_IDS;
        float* out_disc  = out + 2 * L_IDS + C_CODES;
        const size_t cbase = ((size_t)b * NF_ + fbase) * 9;     // 16B-aligned in bytes
        const uint32_t* sc32 = reinterpret_cast<const uint32_t*>(sc);
        const int n4 = FACES_PER_BLOCK * 9 / 4;                 // 4608
        for (int i = tid; i < n4; i += THREADS) {
            uint32_t p = sc32[i];
            float4 val = make_float4((float)(p & 0xFF), (float)((p >> 8) & 0xFF),
                                     (float)((p >> 16) & 0xFF), (float)(p >> 24));
            reinterpret_cast<float4*>(out_codes + cbase)[i] = val;
            reinterpret_cast<float4*>(out_disc  + cbase)[i] = val;
        }
    }

    // Phase B2: input_ids tokens (9 codes + separator per face), contiguous b32 stores.
    {
        const size_t rowbase = (size_t)b * ROW_ + 1 + (size_t)fbase * 10;
        const int np = FACES_PER_BLOCK * 10;                    // 20480
        // last face of batch drops its trailing separator (that slot is the pad)
        const int skip = (fbase + FACES_PER_BLOCK == NF_) ? (np - 1) : -1;
        for (int p = tid; p < np; p += THREADS) {
            int f = p / 10;
            int s = p - f * 10;
            float val = (s == 9) ? 128.0f : (float)sc[f * 9 + s];
            if (p != skip) out[rowbase + p] = val;
        }
    }
}

// ---------------- kernel 3: recon_faces (undiscretize first face per batch) ----------------
__global__ __launch_bounds__(THREADS) void mesh_recon_kernel(
        const float* __restrict__ vertices,
        const int*   __restrict__ faces,
        float*       __restrict__ out)
{
    int t = blockIdx.x * blockDim.x + threadIdx.x;
    if (t >= B_ * 9) return;
    int b = t / 9, k = t - b * 9, v = k / 3, c = k - v * 3;
    int vidx = faces[(size_t)b * NF_ * 3 + v];                  // face 0, vertex v
    float x  = vertices[(size_t)b * NV_ * 3 + (size_t)vidx * 3 + c];
    float q  = quantize(x);
    // undiscretize: (q+0.5)/128*2 - 1
    out[2 * L_IDS + 2 * C_CODES + t] = q * 0.015625f - 0.9921875f;
}

// ---------------- host launch ----------------
extern "C" void kernel_launch(void* const* d_in, const int* in_sizes, int n_in,
                              void* d_out, int out_size, void* d_ws, size_t ws_size,
                              hipStream_t stream) {
    (void)in_sizes; (void)n_in; (void)out_size; (void)d_ws; (void)ws_size;
    const float* vertices = (const float*)d_in[0];
    const int*   faces    = (const int*)d_in[1];
    float*       out      = (float*)d_out;

    // Allow >64KB dynamic LDS (CDNA5 WGP has 320KB); not a stream op, capture-safe.
    (void)hipFuncSetAttribute(reinterpret_cast<const void*>(mesh_tokenize_kernel),
                              hipFuncAttributeMaxDynamicSharedMemorySize, SH_BYTES);

    {   // attention_mask + pads
        size_t n4 = L_IDS / 4;
        int grid = (int)((n4 + THREADS - 1) / THREADS);
        mesh_fill_mask_pads<<<grid, THREADS, 0, stream>>>(out);
    }
    mesh_tokenize_kernel<<<B_ * TILES, THREADS, SH_BYTES, stream>>>(vertices, faces, out);
    mesh_recon_kernel<<<(B_ * 9 + THREADS - 1) / THREADS, THREADS, 0, stream>>>(vertices, faces, out);
}